// DistogramHead_18915035972404
// MI455X (gfx1250) — compile-verified
//
#include <hip/hip_runtime.h>

typedef __attribute__((ext_vector_type(2))) float v2f;
typedef __attribute__((ext_vector_type(4))) float v4f;
typedef __attribute__((ext_vector_type(8))) float v8f;

#define D_MODEL 256
#define D_HALF  128
#define SEQ_L   4096
#define BL      16384   // B*L = 4*4096

// ws layout (floats):
//   [0,256)            v_i = w_i @ w_out
//   [256,512)          v_j = w_j @ w_out
//   [512], [513]       c_i = b_i . w_out ; c_j = b_j . w_out
//   [768,1024)         zeros (B-matrix padding columns 2..15)
//   [1024, 1024+BL)    t_i[g] = 0.5*(s_i+c_i) + b_out
//   [1024+BL, +2*BL)   t_j[g] = 0.5*(s_j+c_j)

__global__ void __launch_bounds__(256)
fold_weights(const float* __restrict__ w_i, const float* __restrict__ b_i,
             const float* __restrict__ w_j, const float* __restrict__ b_j,
             const float* __restrict__ w_out, float* __restrict__ ws) {
  int d = threadIdx.x;                       // 0..255
  float vi = 0.f, vj = 0.f;
  #pragma unroll 8
  for (int h = 0; h < D_HALF; ++h) {
    float wo = w_out[h];
    vi = fmaf(w_i[d * D_HALF + h], wo, vi);
    vj = fmaf(w_j[d * D_HALF + h], wo, vj);
  }
  ws[d]           = vi;
  ws[D_MODEL + d] = vj;
  ws[768 + d]     = 0.f;                     // zero pad-column source
  if (d == 0) {
    float ci = 0.f, cj = 0.f;
    for (int h = 0; h < D_HALF; ++h) {
      ci = fmaf(b_i[h], w_out[h], ci);
      cj = fmaf(b_j[h], w_out[h], cj);
    }
    ws[512] = ci;
    ws[513] = cj;
  }
}

// One wave (32 lanes) per 16-row tile of x. Computes the (BL x 256) @ (256 x 16)
// GEMM with V_WMMA_F32_16X16X4_F32, where B-matrix columns {0,1} = {v_i, v_j}
// and columns 2..15 read from a zero buffer (branch-free per-lane base pointer).
__global__ void __launch_bounds__(32)
scores_wmma(const float* __restrict__ x, const float* __restrict__ b_out_p,
            float* __restrict__ ws) {
  const float* vi    = ws;
  const float* vj    = ws + D_MODEL;
  const float* zeros = ws + 768;
  float* ti = ws + 1024;
  float* tj = ws + 1024 + BL;

  const int lane = threadIdx.x;      // 0..31
  const int m    = lane & 15;        // M index (A) / N index (B,C,D)
  const int half = lane >> 4;        // 0: K={0,1}; 1: K={2,3} within a step
  const int row0 = blockIdx.x * 16;
  const float* xrow = x + (size_t)(row0 + m) * D_MODEL;

  // Per-lane column pointer for the B operand (selected once; no loop branches).
  const float* bptr = (m == 0) ? vi : ((m == 1) ? vj : zeros);

  v8f c = {};
  #pragma unroll 4
  for (int k0 = 0; k0 < D_MODEL; k0 += 4) {
    const int kb = k0 + half * 2;
    // A operand: 16x4 f32 tile (lanes 0-15: K=kb..kb+1 with kb=k0; lanes 16-31: k0+2)
    v2f a = *(const v2f*)(xrow + kb);
    // B operand: rows K=kb,kb+1 of the 256x16 matrix, column n = lane&15.
    v2f b = *(const v2f*)(bptr + kb);
    c = __builtin_amdgcn_wmma_f32_16x16x4_f32(
        /*neg_a=*/false, a, /*neg_b=*/false, b,
        /*c_mod=*/(short)0, c, /*reuse_a=*/false, /*reuse_b=*/false);
  }

  const float ci = ws[512];
  const float cj = ws[513];
  const float bo = b_out_p[0];

  // D layout: VGPR r holds M = r (lanes 0-15) / r+8 (lanes 16-31), N = lane&15.
  if (m == 0) {
    #pragma unroll
    for (int r = 0; r < 8; ++r) {
      const int g = row0 + half * 8 + r;
      ti[g] = 0.5f * (c[r] + ci) + bo;
    }
  } else if (m == 1) {
    #pragma unroll
    for (int r = 0; r < 8; ++r) {
      const int g = row0 + half * 8 + r;
      tj[g] = 0.5f * (c[r] + cj);
    }
  }
}

// out[b,i,j] = relu(t_i[b,i] + t_j[b,j]); 268 MB stream -> non-temporal b128 stores.
__global__ void __launch_bounds__(256)
pair_store(const float* __restrict__ ws, float* __restrict__ out) {
  const float* ti = ws + 1024;
  const float* tj = ws + 1024 + BL;
  const int g = blockIdx.x;          // g = b*L + i
  const int b = g >> 12;             // L = 4096 = 2^12
  const float t = ti[g];
  const float* tjb = tj + ((size_t)b << 12);
  float* orow = out + (size_t)g * SEQ_L;

  #pragma unroll 4
  for (int j = threadIdx.x * 4; j < SEQ_L; j += 256 * 4) {
    v4f v = *(const v4f*)(tjb + j);
    v4f r;
    r.x = fmaxf(t + v.x, 0.f);
    r.y = fmaxf(t + v.y, 0.f);
    r.z = fmaxf(t + v.z, 0.f);
    r.w = fmaxf(t + v.w, 0.f);
    __builtin_nontemporal_store(r, (v4f*)(orow + j));
  }
}

extern "C" void kernel_launch(void* const* d_in, const int* in_sizes, int n_in,
                              void* d_out, int out_size, void* d_ws, size_t ws_size,
                              hipStream_t stream) {
  const float* x     = (const float*)d_in[0];
  const float* w_i   = (const float*)d_in[1];
  const float* b_i   = (const float*)d_in[2];
  const float* w_j   = (const float*)d_in[3];
  const float* b_j   = (const float*)d_in[4];
  const float* w_out = (const float*)d_in[5];
  const float* b_out = (const float*)d_in[6];
  float* ws  = (float*)d_ws;
  float* out = (float*)d_out;

  fold_weights<<<1, 256, 0, stream>>>(w_i, b_i, w_j, b_j, w_out, ws);
  scores_wmma<<<BL / 16, 32, 0, stream>>>(x, b_out, ws);
  pair_store<<<BL, 256, 0, stream>>>(ws, out);
}